// Model_17145509445951
// MI455X (gfx1250) — compile-verified
//
#include <hip/hip_runtime.h>

// Soft Smith-Waterman, anti-diagonal wavefront, one workgroup per batch.
// B=8, M=N=2048, t=1, gap=-1.  Output: per-batch max(H, 0)  (8 floats).
//
// CDNA5 specifics used: wave32 ds_bpermute shuffles, double-buffered LDS
// boundary exchange with SPLIT workgroup barriers (s_barrier_signal -1 /
// s_barrier_wait -1 with work in between), global_prefetch_b8, and
// software-pipelined global loads that stay in flight across the barrier.

#define M_DIM  2048
#define N_DIM  2048
#define NT     512                  // threads per block (16 wave32s)
#define KPT    (N_DIM / NT)         // columns owned per thread = 4
#define NWAVES (NT / 32)            // 16
#define GAP    (-1.0f)
#define PD     16                   // prefetch distance in diagonals
#define DTOT   (M_DIM + N_DIM - 1)  // 4095 diagonals

__device__ __forceinline__ void wg_barrier_signal() {
    // flush our LDS stores, then arrive at the workgroup barrier
    asm volatile("s_wait_dscnt 0x0\n\ts_barrier_signal -1" ::: "memory");
}
__device__ __forceinline__ void wg_barrier_wait() {
    asm volatile("s_barrier_wait -1" ::: "memory");
}

__global__ __launch_bounds__(NT, 1)
void soft_sw_wavefront(const float* __restrict__ S, float* __restrict__ out)
{
    const int b    = blockIdx.x;
    const int tid  = threadIdx.x;
    const int lane = tid & 31;
    const int wave = tid >> 5;
    const float* __restrict__ Sb = S + (size_t)b * (size_t)M_DIM * (size_t)N_DIM;

    // Double-buffered wave-boundary exchange (lane31 -> next wave's lane0).
    __shared__ float bnd1[2][NWAVES][KPT];
    __shared__ float bnd2[2][NWAVES][KPT];
    __shared__ float red[NWAVES];

    if (tid < 2 * NWAVES * KPT) {
        (&bnd1[0][0][0])[tid] = 0.0f;
        (&bnd2[0][0][0])[tid] = 0.0f;
    }

    // h1[k] = H on previous diagonal for column j = tid + NT*k
    // h2[k] = H two diagonals back for that column. Boundary rows/cols are 0.
    float h1[KPT], h2[KPT];
    #pragma unroll
    for (int k = 0; k < KPT; ++k) { h1[k] = 0.0f; h2[k] = 0.0f; }

    float vmax = 0.0f;

    // One diagonal step. 'sv' holds this diagonal's scores (already loaded);
    // loads for the NEXT diagonal are issued first into 'svn' so their
    // latency overlaps the barrier round-trip and the LSE compute.
    auto step = [&](int d, float (&sv)[KPT], float (&svn)[KPT]) {
        const int rbuf = d & 1;
        const int wbuf = rbuf ^ 1;

        // Issue next-diagonal score loads (in flight across the barrier).
        #pragma unroll
        for (int k = 0; k < KPT; ++k) {
            const int j = tid + NT * k;
            const int i = d + 1 - j;
            svn[k] = (i >= 0 && i < M_DIM)
                   ? Sb[((size_t)(unsigned)i << 11) + (unsigned)j] : 0.0f;
        }

        // gfx1250 prefetch: line-leading threads fetch the cache lines that
        // will be first-touched PD diagonals from now.
        if ((tid & 15) == 0) {
            #pragma unroll
            for (int k = 0; k < KPT; ++k) {
                const int j = tid + NT * k;
                const int i = d + PD - j;
                if (i >= 0 && i < M_DIM)
                    __builtin_prefetch(&Sb[((size_t)(unsigned)i << 11) + (unsigned)j], 0, 3);
            }
        }

        // Neighbor column (j-1) state from lane-1 (wave32 shuffle).
        float n1[KPT], n2[KPT];
        #pragma unroll
        for (int k = 0; k < KPT; ++k) {
            n1[k] = __shfl_up(h1[k], 1, 32);
            n2[k] = __shfl_up(h2[k], 1, 32);
        }

        // Wait for the previous round's boundary publication.
        wg_barrier_wait();

        if (lane == 0) {
            #pragma unroll
            for (int k = 0; k < KPT; ++k) {
                int sw = wave - 1, sk = k;
                if (wave == 0) { sw = NWAVES - 1; sk = k - 1; }
                if (sk < 0) { n1[k] = 0.0f; n2[k] = 0.0f; }   // column -1 => 0
                else        { n1[k] = bnd1[rbuf][sw][sk];
                              n2[k] = bnd2[rbuf][sw][sk]; }
            }
        }

        // One stable logsumexp4 per valid cell.
        #pragma unroll
        for (int k = 0; k < KPT; ++k) {
            const int j = tid + NT * k;
            const int i = d - j;
            if (i >= 0 && i < M_DIM) {
                const float a = n2[k] + sv[k];   // H[i-1,j-1] + s
                const float u = h1[k] + GAP;     // H[i-1,j]   + g
                const float l = n1[k] + GAP;     // H[i,j-1]   + g
                const float m = fmaxf(fmaxf(a, u), fmaxf(l, 0.0f));
                const float e = __expf(a - m) + __expf(u - m) +
                                __expf(l - m) + __expf(-m);
                const float h = m + __logf(e);
                vmax  = fmaxf(vmax, h);
                h2[k] = h1[k];
                h1[k] = h;
            }
        }

        // Publish wave-boundary state, then arrive (split barrier: the next
        // iteration's loads/shuffles run before anyone has to wait).
        if (lane == 31) {
            #pragma unroll
            for (int k = 0; k < KPT; ++k) {
                bnd1[wbuf][wave][k] = h1[k];
                bnd2[wbuf][wave][k] = h2[k];
            }
        }
        wg_barrier_signal();
    };

    // Preload scores for diagonal 0.
    float svA[KPT], svB[KPT];
    #pragma unroll
    for (int k = 0; k < KPT; ++k) {
        const int j = tid + NT * k;
        const int i = 0 - j;
        svA[k] = (i >= 0 && i < M_DIM)
               ? Sb[((size_t)(unsigned)i << 11) + (unsigned)j] : 0.0f;
    }

    // Publish the zero-initialized boundary buffers (round 0 arrive).
    wg_barrier_signal();

    // 4095 diagonals = 2047 ping-pong pairs + 1 tail (its d+1 loads all
    // predicate away since row index >= M).
    for (int d = 0; d + 1 < DTOT; d += 2) {
        step(d,     svA, svB);
        step(d + 1, svB, svA);
    }
    step(DTOT - 1, svA, svB);

    // Consume the final arrive so barriers stay balanced.
    wg_barrier_wait();

    // Per-batch max reduction (wave32 shuffle tree + LDS across waves).
    #pragma unroll
    for (int off = 16; off > 0; off >>= 1)
        vmax = fmaxf(vmax, __shfl_xor(vmax, off, 32));
    if (lane == 0) red[wave] = vmax;
    __syncthreads();
    if (wave == 0) {
        float v = (lane < NWAVES) ? red[lane] : 0.0f;
        #pragma unroll
        for (int off = 16; off > 0; off >>= 1)
            v = fmaxf(v, __shfl_xor(v, off, 32));
        if (lane == 0) out[b] = fmaxf(v, 0.0f);
    }
}

extern "C" void kernel_launch(void* const* d_in, const int* in_sizes, int n_in,
                              void* d_out, int out_size, void* d_ws, size_t ws_size,
                              hipStream_t stream)
{
    (void)n_in; (void)d_ws; (void)ws_size; (void)out_size;
    const float* S   = (const float*)d_in[0];
    float*       out = (float*)d_out;
    const int    Bn  = in_sizes[0] / (M_DIM * N_DIM);   // = 8
    soft_sw_wavefront<<<dim3(Bn), dim3(NT), 0, stream>>>(S, out);
}